// HeteroGlobalAttentionPooling_3659312136369
// MI455X (gfx1250) — compile-verified
//
#include <hip/hip_runtime.h>

#define NNODES 100000
#define TT 16
#define DD 128
#define BB 64
#define RR (NNODES * TT)

typedef __attribute__((ext_vector_type(2))) float v2f;
typedef __attribute__((ext_vector_type(8))) float v8f;
typedef __attribute__((ext_vector_type(4))) unsigned int u32x4;
typedef __attribute__((ext_vector_type(4))) int i32x4;
typedef __attribute__((ext_vector_type(8))) int i32x8;

#if __has_builtin(__builtin_amdgcn_tensor_load_to_lds) && \
    __has_builtin(__builtin_amdgcn_s_wait_tensorcnt)
#define USE_TDM 1
#else
#define USE_TDM 0
#endif

// ---------------------------------------------------------------------------
// Kernel 0: zero the output accumulator and softmax statistics.
// ---------------------------------------------------------------------------
__global__ __launch_bounds__(256) void init_kernel(float* __restrict__ out,
                                                   float* __restrict__ segsum,
                                                   unsigned* __restrict__ segmax) {
    int i = blockIdx.x * blockDim.x + threadIdx.x;
    if (i < BB * DD) out[i] = 0.0f;
    if (i < BB * TT) { segsum[i] = 0.0f; segmax[i] = 0u; }
}

// order-preserving float -> uint mapping for atomic max
__device__ inline unsigned fkey(float f) {
    int i = __float_as_int(f);
    return (i >= 0) ? (unsigned(i) | 0x80000000u) : ~unsigned(i);
}
__device__ inline float funkey(unsigned u) {
    int i = (u & 0x80000000u) ? int(u ^ 0x80000000u) : ~int(u);
    return __int_as_float(i);
}

#if USE_TDM
// Tensor Data Mover: issue (no wait) a DMA of one 16x128 f32 tile global->LDS,
// with HW padding of 4 DWORDs after every 128 DWORDs so the LDS row stride is
// 132 floats (bank-conflict-free for the WMMA A-pattern reads).
__device__ inline void tdm_issue_tile(const float* gsrc, float* lds_tile) {
    unsigned long long ga = (unsigned long long)(size_t)gsrc;
    unsigned lofs =
        (unsigned)(size_t)(__attribute__((address_space(3))) float*)lds_tile;

    u32x4 g0;
    g0.x = 1u;                                     // count=1 (valid descriptor)
    g0.y = lofs;                                   // lds_addr (bytes)
    g0.z = (unsigned)(ga & 0xFFFFFFFFu);           // global_addr[31:0]
    g0.w = (unsigned)((ga >> 32) & 0x01FFFFFFu)    // global_addr[56:32]
         | (2u << 30);                             // type = 2 ("image")

    i32x8 g1;
    g1[0] = (2 << 16)      // data_size = 4 bytes
          | (1 << 20)      // pad_enable
          | (6 << 22)      // pad_interval: every 128 DWORDs
          | (3 << 25);     // pad_amount: 4 DWORDs
    g1[1] = (int)(128u << 16);   // tensor_dim0 = 128 (bits 79:48, low half)
    g1[2] = (int)(16u << 16);    // tensor_dim1 = 16  (bits 111:80, low half)
    g1[3] = (int)(128u << 16);   // tile_dim0 = 128   (bits 127:112)
    g1[4] = 16;                  // tile_dim1 = 16, tile_dim2 = 0
    g1[5] = 128;                 // tensor_dim0_stride = 128 (low 32 of 48)
    g1[6] = 0;
    g1[7] = 0;

    i32x4 z4 = {0, 0, 0, 0};
#if defined(__clang_major__) && __clang_major__ >= 23
    i32x8 z8 = {0, 0, 0, 0, 0, 0, 0, 0};
    __builtin_amdgcn_tensor_load_to_lds(g0, g1, z4, z4, z8, 0);
#else
    __builtin_amdgcn_tensor_load_to_lds(g0, g1, z4, z4, 0);
#endif
}
#endif

// ---------------------------------------------------------------------------
// Kernel 1: gate logits via V_WMMA_F32_16X16X4_F32 + per-(graph,t) atomic max.
// One wave owns one node: A-tile = feat[n] (16 t-rows x 128 k), chained over
// K in steps of 4; Wg lives in B column 0 so C[:,0] = gate logits for t=0..15.
// TDM double-buffered: DMA of node n+stride overlaps WMMA on node n.
// 2 waves/block so 2 buffers x 8448B x 2 waves stays under the static-LDS cap.
// ---------------------------------------------------------------------------
#define GWPB 2   // waves per block in gate kernel
__global__ __launch_bounds__(32 * GWPB) void gate_max_kernel(
        const float* __restrict__ feat, const float* __restrict__ Wg,
        const float* __restrict__ bg, const int* __restrict__ seg,
        float* __restrict__ gate, unsigned* __restrict__ segmax) {
    __shared__ float lwg[DD];
    __shared__ float tile[GWPB][2][16 * 132];  // double-buffered padded tiles
    __shared__ float gout[GWPB][16];

    int tid = threadIdx.x;
    for (int i = tid; i < DD; i += 32 * GWPB) lwg[i] = Wg[i];
    __syncthreads();

    int w = tid >> 5, lane = tid & 31;
    int row  = lane & 15;                    // A-matrix M index held by this lane
    int koff = (lane >> 4) * 2;              // lanes 16..31 hold K=2,3 of each chunk
    float bmask = (row == 0) ? 1.0f : 0.0f;  // B column 0 only; no EXEC churn
    float bgv = bg[0];

    int gw = blockIdx.x * GWPB + w;
    int stride = gridDim.x * GWPB;
    int buf = 0;

#if USE_TDM
    if (gw < NNODES)
        tdm_issue_tile(feat + (size_t)gw * (TT * DD), tile[w][0]);
#endif

    for (int n = gw; n < NNODES; n += stride) {
        float* tl = tile[w][buf];
        int nn = n + stride;
#if USE_TDM
        if (nn < NNODES) {
            // drain any in-flight LDS reads of the alternate buffer (WAR)
            asm volatile("s_wait_dscnt 0x0" ::: "memory");
            tdm_issue_tile(feat + (size_t)nn * (TT * DD), tile[w][buf ^ 1]);
            __builtin_amdgcn_s_wait_tensorcnt((short)1);  // current tile done
        } else {
            __builtin_amdgcn_s_wait_tensorcnt((short)0);
        }
#else
        const float* src = feat + (size_t)n * (TT * DD);
        #pragma unroll
        for (int i = 0; i < 16; ++i)
            *(float4*)(tl + i * 132 + lane * 4) =
                *(const float4*)(src + i * DD + lane * 4);
#endif
        __builtin_amdgcn_wave_barrier();

        v8f c = {0.f, 0.f, 0.f, 0.f, 0.f, 0.f, 0.f, 0.f};
        #pragma unroll
        for (int kk = 0; kk < DD; kk += 4) {
            // A: lane holds row (lane&15), K = kk + koff + {0,1}
            v2f a = *(const v2f*)(tl + row * 132 + kk + koff);
            // B: unconditional load + mask-mul (only lane0/lane16 contribute)
            v2f wgk = *(const v2f*)(lwg + kk + koff);
            v2f b = { wgk.x * bmask, wgk.y * bmask };
            c = __builtin_amdgcn_wmma_f32_16x16x4_f32(false, a, false, b,
                                                      (short)0, c, false, false);
        }
        // C[:,0]: m=0..7 -> VGPR m lane 0; m=8..15 -> VGPR m-8 lane 16
        if (lane == 0) {
            #pragma unroll
            for (int j = 0; j < 8; ++j) gout[w][j] = c[j];
        }
        if (lane == 16) {
            #pragma unroll
            for (int j = 0; j < 8; ++j) gout[w][8 + j] = c[j];
        }
        __builtin_amdgcn_wave_barrier();
        if (lane < TT) {
            float g = gout[w][lane] + bgv;
            gate[(size_t)n * TT + lane] = g;
            int s = seg[n];
            atomicMax(&segmax[s * TT + lane], fkey(g));
        }
        __builtin_amdgcn_wave_barrier();
        buf ^= 1;
    }
}

// ---------------------------------------------------------------------------
// Kernel 2: e = exp(g - max); in-place into gate scratch; atomic segment sum.
// ---------------------------------------------------------------------------
__global__ __launch_bounds__(256) void exp_sum_kernel(const int* __restrict__ seg,
                                                      float* __restrict__ gate,
                                                      const unsigned* __restrict__ segmax,
                                                      float* __restrict__ segsum) {
    int r = blockIdx.x * blockDim.x + threadIdx.x;
    if (r >= RR) return;
    int n = r >> 4, t = r & 15;
    int s = seg[n];
    float m = funkey(segmax[s * TT + t]);
    float e = __expf(gate[r] - m);
    gate[r] = e;
    atomicAdd(&segsum[s * TT + t], e);
}

// ---------------------------------------------------------------------------
// Kernel 3: z -> 1/z (1024 scalars).
// ---------------------------------------------------------------------------
__global__ __launch_bounds__(256) void invz_kernel(float* __restrict__ segsum) {
    int i = blockIdx.x * blockDim.x + threadIdx.x;
    if (i < BB * TT) segsum[i] = 1.0f / segsum[i];
}

// ---------------------------------------------------------------------------
// Kernel 4: readout[b,d] += sum_t w[n,t]*feat[n,t,d]. Each wave owns a
// contiguous node range (segment_ids are sorted), lane owns 4 d-values in
// registers; atomic flush only on segment change.
// ---------------------------------------------------------------------------
__global__ __launch_bounds__(256) void readout_kernel(const float* __restrict__ feat,
                                                      const int* __restrict__ seg,
                                                      const float* __restrict__ e,
                                                      const float* __restrict__ invz,
                                                      float* __restrict__ out) {
    int gtid  = blockIdx.x * blockDim.x + threadIdx.x;
    int wave  = gtid >> 5;
    int lane  = threadIdx.x & 31;
    int nwaves = (gridDim.x * blockDim.x) >> 5;
    int per = (NNODES + nwaves - 1) / nwaves;
    int n0 = wave * per;
    int n1 = n0 + per; if (n1 > NNODES) n1 = NNODES;
    if (n0 >= n1) return;

    float ax = 0.f, ay = 0.f, az = 0.f, aw = 0.f;
    int cur = seg[n0];

    for (int n = n0; n < n1; ++n) {
        int s = seg[n];
        if (s != cur) {
            float* o = out + cur * DD + lane * 4;
            atomicAdd(o + 0, ax); atomicAdd(o + 1, ay);
            atomicAdd(o + 2, az); atomicAdd(o + 3, aw);
            ax = ay = az = aw = 0.f;
            cur = s;
        }
        const float* frow = feat + (size_t)n * (TT * DD);
        if (n + 1 < n1)  // pull next node's rows toward the caches
            __builtin_prefetch(feat + (size_t)(n + 1) * (TT * DD) + lane * 16, 0, 0);
        #pragma unroll
        for (int t = 0; t < TT; ++t) {
            float wgt = e[(size_t)n * TT + t] * invz[cur * TT + t];
            const float4 x = *(const float4*)(frow + t * DD + lane * 4);
            ax += wgt * x.x; ay += wgt * x.y;
            az += wgt * x.z; aw += wgt * x.w;
        }
    }
    float* o = out + cur * DD + lane * 4;
    atomicAdd(o + 0, ax); atomicAdd(o + 1, ay);
    atomicAdd(o + 2, az); atomicAdd(o + 3, aw);
}

// ---------------------------------------------------------------------------
extern "C" void kernel_launch(void* const* d_in, const int* in_sizes, int n_in,
                              void* d_out, int out_size, void* d_ws, size_t ws_size,
                              hipStream_t stream) {
    const float* feat = (const float*)d_in[0];
    const float* Wg   = (const float*)d_in[1];
    const float* bg   = (const float*)d_in[2];
    const int*   seg  = (const int*)d_in[3];
    float* out = (float*)d_out;

    float*    gate   = (float*)d_ws;                                   // RR floats
    unsigned* segmax = (unsigned*)((char*)d_ws + (size_t)RR * 4);      // BB*TT u32
    float*    segsum = (float*)((char*)segmax + (size_t)BB * TT * 4);  // BB*TT f32

    init_kernel<<<(BB * DD + 255) / 256, 256, 0, stream>>>(out, segsum, segmax);
    gate_max_kernel<<<1024, 32 * GWPB, 0, stream>>>(feat, Wg, bg, seg, gate, segmax);
    exp_sum_kernel<<<(RR + 255) / 256, 256, 0, stream>>>(seg, gate, segmax, segsum);
    invz_kernel<<<(BB * TT + 255) / 256, 256, 0, stream>>>(segsum);
    readout_kernel<<<128, 256, 0, stream>>>(feat, seg, gate, segsum, out);
}